// PillarScatter_81252191306133
// MI455X (gfx1250) — compile-verified
//
#include <hip/hip_runtime.h>
#include <stdint.h>

// PointPillars BEV scatter for MI455X (gfx1250).
// out (B=4, C=64, H=496, W=432) NCHW f32; feat (M,64) f32; coords (M,3) i32.

#define BEV_H 496
#define BEV_W 432
#define CCH   64
#define PLANE (BEV_H * BEV_W)   // 214272

// int4 pointer types for the async-to-LDS builtin (params are vector-of-4-int ptrs)
typedef int v4i __attribute__((ext_vector_type(4)));
typedef __attribute__((address_space(1))) v4i gbl_v4i;   // global (AS1)
typedef __attribute__((address_space(3))) v4i lds_v4i;   // LDS (AS3)

// ---------------------------------------------------------------------------
// Kernel A: zero-fill the canvas with b128 stores (bulk of the 219 MB traffic)
// ---------------------------------------------------------------------------
__global__ void PillarScatter_fill_kernel(float4* __restrict__ out, int n4) {
    int stride = gridDim.x * blockDim.x;
    float4 z = make_float4(0.f, 0.f, 0.f, 0.f);
    for (int i = blockIdx.x * blockDim.x + threadIdx.x; i < n4; i += stride) {
        out[i] = z;
    }
}

// ---------------------------------------------------------------------------
// Kernel B: scatter. 16 threads per pillar, 4 channels (one float4) each.
// Feature rows are DMA'd global->LDS asynchronously (ASYNCcnt) while coords
// are fetched and the scatter base address is computed, then the wave waits
// on s_wait_asynccnt and issues the 4 strided stores.
// ---------------------------------------------------------------------------
__global__ void PillarScatter_scatter_kernel(const int*   __restrict__ coords,
                                             const float* __restrict__ feats,
                                             float*       __restrict__ out,
                                             int M) {
    __shared__ float4 smem[256];

    int tid    = blockIdx.x * blockDim.x + threadIdx.x;
    int pillar = tid >> 4;          // 16 threads per pillar
    int q      = tid & 15;          // channel-quad index: channels 4q..4q+3

    if (pillar < M) {
        // --- kick off async DMA of this thread's 16B feature quad into LDS ---
        const float* gsrc = feats + (size_t)pillar * CCH + (size_t)q * 4;
#if __has_builtin(__builtin_amdgcn_global_load_async_to_lds_b128)
        __builtin_amdgcn_global_load_async_to_lds_b128(
            (gbl_v4i*)(uintptr_t)gsrc,
            (lds_v4i*)(uintptr_t)&smem[threadIdx.x],
            /*offset=*/0, /*cpol=*/0);
#else
        {
            unsigned lds_off = (unsigned)(uintptr_t)&smem[threadIdx.x];
            // CDNA5 ISA: VDST = LDS byte addr, VADDR = 64b global addr
            asm volatile("global_load_async_to_lds_b128 %0, %1, off"
                         :: "v"(lds_off), "v"(gsrc)
                         : "memory");
        }
#endif

        // --- overlap: load coords and compute scatter base while DMA is in flight ---
        int bi = coords[pillar * 3 + 0];
        int yi = coords[pillar * 3 + 1];
        int xi = coords[pillar * 3 + 2];
        size_t base = (size_t)(bi * CCH + q * 4) * (size_t)PLANE
                    + (size_t)yi * BEV_W + (size_t)xi;

        // --- wait for the async copy, read back our quad from LDS ---
#if __has_builtin(__builtin_amdgcn_s_wait_asynccnt)
        __builtin_amdgcn_s_wait_asynccnt(0);
#else
        asm volatile("s_wait_asynccnt 0x0" ::: "memory");
#endif
        asm volatile("" ::: "memory");   // keep the LDS read after the wait
        float4 f = smem[threadIdx.x];

        // --- 4 channel-plane stores, stride H*W elements ---
        out[base]                     = f.x;
        out[base + (size_t)PLANE]     = f.y;
        out[base + (size_t)2 * PLANE] = f.z;
        out[base + (size_t)3 * PLANE] = f.w;
    }
}

extern "C" void kernel_launch(void* const* d_in, const int* in_sizes, int n_in,
                              void* d_out, int out_size, void* d_ws, size_t ws_size,
                              hipStream_t stream) {
    const int*   coords = (const int*)d_in[0];    // (M,3) int32
    const float* feats  = (const float*)d_in[1];  // (M,64) float32
    float*       out    = (float*)d_out;          // (B,64,496,432) float32, flat

    int M  = in_sizes[0] / 3;
    int n4 = out_size / 4;                        // out_size divisible by 4

    // Zero-fill: grid-stride, 8 waves/block, plenty of blocks to saturate HBM.
    PillarScatter_fill_kernel<<<4096, 256, 0, stream>>>((float4*)out, n4);

    // Scatter: 16 threads per pillar.
    long long threads = (long long)M * 16;
    int blocks = (int)((threads + 255) / 256);
    PillarScatter_scatter_kernel<<<blocks, 256, 0, stream>>>(coords, feats, out, M);
}